// GLM_Encoder_54082228191293
// MI455X (gfx1250) — compile-verified
//
#include <hip/hip_runtime.h>
#include <hip/hip_bf16.h>
#include <math.h>

// Problem constants (fixed by setup_inputs in the reference)
#define E_NO   800
#define I_NO   200
#define SUB1   15      // sub_no - 1
#define TNO    100
#define COSN   13
#define OBSN   (2*COSN - 1)   // 25
#define OBSK   (2*TNO + 1)    // 201
#define KB_E   (E_NO/4)       // 200
#define KB_I   (I_NO/4)       // 50

typedef __attribute__((ext_vector_type(2))) float v2f;
typedef __attribute__((ext_vector_type(8))) float v8f;

// ---------------------------------------------------------------------------
// Kernel 1: build filter banks + pack C matrices for WMMA (runs as 1 block)
// ---------------------------------------------------------------------------
__global__ void glm_prep_kernel(const float* __restrict__ CsE,    // [16,800]
                                const float* __restrict__ CsI,    // [16,200]
                                const float* __restrict__ Wsyn,   // [15,13,2]
                                const float* __restrict__ Whist,  // [15,13]
                                const float* __restrict__ Wobs,   // [15,25]
                                float* __restrict__ CeP,          // [200][16][4]
                                float* __restrict__ CiP,          // [50][16][4]
                                float* __restrict__ ek,           // [15][100]
                                float* __restrict__ ik,           // [15][100]
                                float* __restrict__ hk,           // [15][100]
                                float* __restrict__ ok)           // [15][201]
{
    __shared__ float cb[COSN * TNO];     // raised-cosine log-time basis
    __shared__ float ob[OBSN * OBSK];    // acausal observation basis
    const int tid = threadIdx.x;
    const double PI = 3.14159265358979323846;

    // cos_basis[c][d]
    for (int i = tid; i < COSN * TNO; i += blockDim.x) {
        int c = i / TNO, d = i % TNO;
        double raw = 5.0 * log((double)d + 1.0 + 1e-8);
        double phi = 1.5707963267948966 * (double)c;
        float v = 0.f;
        if (!(raw < phi - PI || raw > phi + PI))
            v = (float)(0.5 * cos(raw - phi) + 0.5);
        cb[i] = v;
    }
    // obs_basis[c][j], j indexes x = j-100 in [-100,100]
    for (int i = tid; i < OBSN * OBSK; i += blockDim.x) {
        int c = i / OBSK, j = i % OBSK;
        float v = 0.f;
        if (c == 0) {
            double xo = (double)(j - TNO);
            double ro = 5.0 * log(fabs(xo) + 1.0 + 1e-8);
            if (!(ro < -PI || ro > PI)) v = (float)(0.5 * cos(ro) + 0.5);
        } else {
            int i2 = (c + 1) / 2;                 // 1..12
            double phi = 1.5707963267948966 * (double)i2;
            bool pos = (c & 1);                   // odd c -> positive side row
            if (pos && j >= TNO) {
                double xp = (double)(j - TNO);
                double rp = 5.0 * log(xp + 1.0 + 1e-8);
                if (!(rp < phi - PI || rp > phi + PI))
                    v = (float)(0.5 * cos(rp - phi) + 0.5);
            } else if (!pos && j <= TNO) {
                double xn = (double)(j - TNO);    // -100..0
                double rn = 5.0 * log(fabs(xn) + 1.0 + 1e-8);
                if (!(rn < phi - PI || rn > phi + PI))
                    v = (float)(0.5 * cos(rn - phi) + 0.5);
            }
        }
        ob[i] = v;
    }
    __syncthreads();

    // Contract weights with bases (unflipped = direct correlation kernels)
    for (int i = tid; i < SUB1 * TNO; i += blockDim.x) {
        int s = i / TNO, d = i % TNO;
        float ae = 0.f, ai = 0.f, ah = 0.f;
        for (int c = 0; c < COSN; ++c) {
            float b = cb[c * TNO + d];
            ae += Wsyn[(s * COSN + c) * 2 + 0] * b;
            ai += Wsyn[(s * COSN + c) * 2 + 1] * b;
            ah += Whist[s * COSN + c] * b;
        }
        ek[i] = ae; ik[i] = ai; hk[i] = ah;
    }
    for (int i = tid; i < SUB1 * OBSK; i += blockDim.x) {
        int s = i / OBSK, m = i % OBSK;
        float a = 0.f;
        for (int c = 0; c < OBSN; ++c) a += Wobs[s * OBSN + c] * ob[c * OBSK + m];
        ok[i] = a;
    }

    // Pack C matrices: CeP[kb][n][kk] = C_syn_e[n+1][kb*4+kk] (n<15, else 0)
    for (int i = tid; i < KB_E * 64; i += blockDim.x) {
        int kb = i >> 6, n = (i >> 2) & 15, kk = i & 3;
        CeP[i] = (n < SUB1) ? CsE[(size_t)(n + 1) * E_NO + kb * 4 + kk] : 0.f;
    }
    for (int i = tid; i < KB_I * 64; i += blockDim.x) {
        int kb = i >> 6, n = (i >> 2) & 15, kk = i & 3;
        CiP[i] = (n < SUB1) ? CsI[(size_t)(n + 1) * I_NO + kb * 4 + kk] : 0.f;
    }
}

// ---------------------------------------------------------------------------
// Kernel 2: WMMA f32 GEMM. One wave32 per 16-timestep tile.
// synE[T][16] = S_e @ CeP ; synI[T][16] = S_i @ CiP
// ---------------------------------------------------------------------------
__global__ void glm_gemm_kernel(const float* __restrict__ Se,
                                const float* __restrict__ Si,
                                const float* __restrict__ CeP,
                                const float* __restrict__ CiP,
                                float* __restrict__ synE,
                                float* __restrict__ synI,
                                int T)
{
    const int wave = (blockIdx.x * blockDim.x + threadIdx.x) >> 5; // wave32
    const int lane = threadIdx.x & 31;
    const int t0 = wave * 16;
    if (t0 >= T) return;                 // wave-uniform: EXEC stays all-1s

    const int half = lane >> 4;          // 0: K{0,1}, 1: K{2,3}
    const int l    = lane & 15;          // A: row, B: column

    // A fragment base: S[t0+l][half*2], contiguous pair per K-block (b64 load)
    const float* arow_e = Se + (size_t)(t0 + l) * E_NO + half * 2;
    const float* arow_i = Si + (size_t)(t0 + l) * I_NO + half * 2;
    // B fragment base: CeP[kb][l][half*2], contiguous pair (b64 load)
    const float* brow_e = CeP + l * 4 + half * 2;
    const float* brow_i = CiP + l * 4 + half * 2;

    v8f accE = {};
    v8f accI = {};

    #pragma unroll 4
    for (int kb = 0; kb < KB_E; ++kb) {
        v2f a = *(const v2f*)(arow_e + kb * 4);
        v2f b = *(const v2f*)(brow_e + kb * 64);
        accE = __builtin_amdgcn_wmma_f32_16x16x4_f32(
            false, a, false, b, (short)0, accE, false, false);
    }
    #pragma unroll 4
    for (int kb = 0; kb < KB_I; ++kb) {
        v2f a = *(const v2f*)(arow_i + kb * 4);
        v2f b = *(const v2f*)(brow_i + kb * 64);
        accI = __builtin_amdgcn_wmma_f32_16x16x4_f32(
            false, a, false, b, (short)0, accI, false, false);
    }

    // D layout: VGPR v -> M = v + half*8, N = l
    #pragma unroll
    for (int v = 0; v < 8; ++v) {
        int m = v + half * 8;
        synE[(size_t)(t0 + m) * 16 + l] = accE[v];
        synI[(size_t)(t0 + m) * 16 + l] = accI[v];
    }
}

// ---------------------------------------------------------------------------
// Kernel 3: all filters + exp, 64 timesteps per block, LDS-staged halos.
//   filt_e[t,s] = sum_d synE[t-d][s]   * ek[s][d]      (d = 0..99)
//   filt_i[t,s] = sum_d synI[t-d][s]   * ik[s][d]
//   hist  [t,s] = sum_d Zh  [t-1-d][s] * hk[s][d]
//   obs   [t,s] = sum_m Zo  [t+100-m]  * ok[s][m]      (m = 0..200)
//   out = exp(filt_e + filt_i + hist + obs + Theta[s]), written twice.
// ---------------------------------------------------------------------------
#define CONV_TT   64
#define SLAB_ROWS (CONV_TT + TNO)   // 164, covers t0-100 .. t0+63

__global__ void glm_conv_kernel(const float* __restrict__ synE,
                                const float* __restrict__ synI,
                                const float* __restrict__ Zh,   // [T][15]
                                const float* __restrict__ Zo,   // [T]
                                const float* __restrict__ ek,
                                const float* __restrict__ ik,
                                const float* __restrict__ hk,
                                const float* __restrict__ ok,
                                const float* __restrict__ Theta,
                                float* __restrict__ out,
                                int T)
{
    __shared__ float sE[SLAB_ROWS * 16];
    __shared__ float sI[SLAB_ROWS * 16];
    __shared__ float sH[SLAB_ROWS * 16];
    __shared__ float sO[CONV_TT + 2 * TNO + 4];   // 268

    const int t0  = blockIdx.x * CONV_TT;
    const int tid = threadIdx.x;

    // Stage slabs: row r <-> time t0 - 100 + r
    for (int idx = tid; idx < SLAB_ROWS * 16; idx += blockDim.x) {
        int r = idx >> 4, s = idx & 15;
        int t = t0 - TNO + r;
        float ve = 0.f, vi = 0.f, vh = 0.f;
        if (t >= 0 && t < T) {
            ve = synE[(size_t)t * 16 + s];
            vi = synI[(size_t)t * 16 + s];
            if (s < SUB1) vh = Zh[(size_t)t * SUB1 + s];
        }
        sE[idx] = ve; sI[idx] = vi; sH[idx] = vh;
    }
    for (int idx = tid; idx < CONV_TT + 2 * TNO; idx += blockDim.x) {
        int t = t0 - TNO + idx;
        sO[idx] = (t >= 0 && t < T) ? Zo[t] : 0.f;
    }
    __syncthreads();

    // 1024 (t,s) cells per block, 4 per thread
    for (int q = 0; q < 4; ++q) {
        int idx = q * 256 + tid;
        int tl = idx >> 4, s = idx & 15;
        if (s >= SUB1) continue;
        int t = t0 + tl;
        if (t >= T) continue;

        float acc = Theta[s];
        const float* eks = ek + s * TNO;
        const float* iks = ik + s * TNO;
        const float* hks = hk + s * TNO;
        #pragma unroll 4
        for (int d = 0; d < TNO; ++d) {
            int r = tl + TNO - d;                 // time t - d
            acc += sE[r * 16 + s] * eks[d];
            acc += sI[r * 16 + s] * iks[d];
            acc += sH[(r - 1) * 16 + s] * hks[d]; // time t - 1 - d
        }
        const float* oks = ok + s * OBSK;
        #pragma unroll 4
        for (int m = 0; m < OBSK; ++m)
            acc += sO[tl + 2 * TNO - m] * oks[m]; // time t + 100 - m

        float o = expf(acc);
        out[(size_t)t * SUB1 + s] = o;
        out[(size_t)T * SUB1 + (size_t)t * SUB1 + s] = o;
    }
}

// ---------------------------------------------------------------------------
// Launch
// ---------------------------------------------------------------------------
extern "C" void kernel_launch(void* const* d_in, const int* in_sizes, int n_in,
                              void* d_out, int out_size, void* d_ws, size_t ws_size,
                              hipStream_t stream)
{
    const float* S_e   = (const float*)d_in[0];
    const float* S_i   = (const float*)d_in[1];
    const float* Z_obs = (const float*)d_in[2];
    const float* Z_hid = (const float*)d_in[3];
    const float* CsE   = (const float*)d_in[4];
    const float* CsI   = (const float*)d_in[5];
    const float* Wsyn  = (const float*)d_in[6];
    const float* Whist = (const float*)d_in[7];
    const float* Wobs  = (const float*)d_in[8];
    const float* Theta = (const float*)d_in[9];

    const int T = in_sizes[2];   // 120000

    // Workspace layout (floats)
    float* ws = (float*)d_ws;
    float* CeP  = ws;                       // 200*64   = 12800
    float* CiP  = CeP + KB_E * 64;          // 50*64    = 3200
    float* ek   = CiP + KB_I * 64;          // 15*100
    float* ik   = ek + SUB1 * TNO;
    float* hk   = ik + SUB1 * TNO;
    float* ok   = hk + SUB1 * TNO;          // 15*201
    float* synE = ok + ((SUB1 * OBSK + 255) & ~255);
    float* synI = synE + (size_t)T * 16;

    float* out = (float*)d_out;

    // 1. Filter banks + packed C matrices
    glm_prep_kernel<<<1, 256, 0, stream>>>(CsE, CsI, Wsyn, Whist, Wobs,
                                           CeP, CiP, ek, ik, hk, ok);

    // 2. WMMA GEMM: one wave per 16 timesteps, 8 waves per block
    int tiles  = (T + 15) / 16;             // 7500
    int blocks = (tiles + 7) / 8;           // 938
    glm_gemm_kernel<<<blocks, 256, 0, stream>>>(S_e, S_i, CeP, CiP, synE, synI, T);

    // 3. Filters + exp
    int cblocks = (T + CONV_TT - 1) / CONV_TT;  // 1875
    glm_conv_kernel<<<cblocks, 256, 0, stream>>>(synE, synI, Z_hid, Z_obs,
                                                 ek, ik, hk, ok, Theta, out, T);
}